// SVDHead_11458972746139
// MI455X (gfx1250) — compile-verified
//
#include <hip/hip_runtime.h>
#include <hip/hip_bf16.h>
#include <math.h>

typedef __attribute__((ext_vector_type(2))) float v2f;
typedef __attribute__((ext_vector_type(8))) float v8f;

#define BATCH 8
#define DIM   512
#define NPTS  2048

// d_out flat layout (floats): Rm[B*9] | T[B*3] | corres[B*N] | weight[B*N]
#define OFF_R 0
#define OFF_T 72
#define OFF_C 96
#define OFF_W (96 + BATCH * NPTS)

__device__ __forceinline__ void sort3f(float& a, float& b, float& c) {
    float t;
    if (a > b) { t = a; a = b; b = t; }
    if (b > c) { t = b; b = c; c = t; }
    if (a > b) { t = a; a = b; b = t; }
}

__device__ __forceinline__ float det3(const float A[3][3]) {
    return A[0][0] * (A[1][1] * A[2][2] - A[1][2] * A[2][1])
         - A[0][1] * (A[1][0] * A[2][2] - A[1][2] * A[2][0])
         + A[0][2] * (A[1][0] * A[2][1] - A[1][1] * A[2][0]);
}

// ---------------------------------------------------------------------------
// Kernel 0: init per-batch loss-min slots to +inf (as uint-ordered float)
// ---------------------------------------------------------------------------
__global__ void init_min_kernel(unsigned int* __restrict__ mn) {
    if (threadIdx.x < BATCH) mn[threadIdx.x] = 0x7F800000u; // +inf
}

// ---------------------------------------------------------------------------
// Kernel 1: fused WMMA score GEMM + online softmax + argmax + corr_tgt.
// One wave per 16-row strip; 8 waves / block; grid = B*128/8 = 128.
// Register blocking: 4 column tiles (64 cols) per pass over K so one A
// fragment feeds 4 v_wmma ops (2 A + 8 B loads per 4 WMMA).
// Branchless online-softmax update (no exec-mask divergence).
// ---------------------------------------------------------------------------
__global__ __launch_bounds__(256) void score_argmax_kernel(
    const float* __restrict__ srcE, const float* __restrict__ tgtE,
    const float* __restrict__ tgt, float* __restrict__ corr,
    int* __restrict__ corres_i, float* __restrict__ out)
{
    const int lane  = threadIdx.x & 31;
    const int wave  = threadIdx.x >> 5;
    const int strip = blockIdx.x * 8 + wave;      // 0 .. B*128-1
    const int b     = strip >> 7;                 // strip / 128
    const int n0    = (strip & 127) << 4;         // row-tile base
    const int l15   = lane & 15;
    const int half  = lane >> 4;

    const float* Ab = srcE + b * DIM * NPTS;
    const float* Bb = tgtE + b * DIM * NPTS;
    const float* tb = tgt  + b * 3   * NPTS;
    const int arow  = n0 + l15;
    const float F   = 441.94173824159216f;        // 10000 / sqrt(512)

    // online-softmax state: row = n0 + half*8 + v, col stream = group cols
    float M[8], S[8], P0[8], P1[8], P2[8];
    int   BI[8];
#pragma unroll
    for (int v = 0; v < 8; ++v) {
        M[v] = -3.4e38f; S[v] = 0.f; P0[v] = 0.f; P1[v] = 0.f; P2[v] = 0.f; BI[v] = 0;
    }

    for (int g = 0; g < NPTS / 64; ++g) {         // 32 groups of 4 col tiles
        v8f acc[4];
#pragma unroll
        for (int t = 0; t < 4; ++t) acc[t] = (v8f){0.f,0.f,0.f,0.f,0.f,0.f,0.f,0.f};

        const float* ap = Ab + (half * 2) * NPTS + arow;
        const float* bp = Bb + (half * 2) * NPTS + g * 64 + l15;
#pragma unroll 2
        for (int k = 0; k < DIM; k += 4) {
            v2f a, b0, b1, b2, b3;
            a.x  = ap[0];        a.y  = ap[NPTS];
            b0.x = bp[0];        b0.y = bp[NPTS];
            b1.x = bp[16];       b1.y = bp[NPTS + 16];
            b2.x = bp[32];       b2.y = bp[NPTS + 32];
            b3.x = bp[48];       b3.y = bp[NPTS + 48];
            ap += 4 * NPTS;      bp += 4 * NPTS;
            acc[0] = __builtin_amdgcn_wmma_f32_16x16x4_f32(
                false, a, false, b0, (short)0, acc[0], false, false);
            acc[1] = __builtin_amdgcn_wmma_f32_16x16x4_f32(
                false, a, false, b1, (short)0, acc[1], false, false);
            acc[2] = __builtin_amdgcn_wmma_f32_16x16x4_f32(
                false, a, false, b2, (short)0, acc[2], false, false);
            acc[3] = __builtin_amdgcn_wmma_f32_16x16x4_f32(
                false, a, false, b3, (short)0, acc[3], false, false);
        }

#pragma unroll
        for (int t = 0; t < 4; ++t) {
            const int col = g * 64 + t * 16 + l15;
            const float t0 = tb[col], t1 = tb[NPTS + col], t2 = tb[2 * NPTS + col];
#pragma unroll
            for (int v = 0; v < 8; ++v) {
                float s  = acc[t][v] * F;
                float nM = fmaxf(M[v], s);
                float e1 = __expf(M[v] - nM);
                float e2 = __expf(s - nM);
                S[v]  = S[v]  * e1 + e2;
                P0[v] = P0[v] * e1 + e2 * t0;
                P1[v] = P1[v] * e1 + e2 * t1;
                P2[v] = P2[v] * e1 + e2 * t2;
                if (s > M[v]) BI[v] = col;        // strict >: keeps first argmax
                M[v] = nM;
            }
        }
    }

    // merge partials across the 16 lanes of each half (cols of the row)
#pragma unroll
    for (int off = 1; off < 16; off <<= 1) {
#pragma unroll
        for (int v = 0; v < 8; ++v) {
            float oM = __shfl_xor(M[v],  off, 32);
            float oS = __shfl_xor(S[v],  off, 32);
            float o0 = __shfl_xor(P0[v], off, 32);
            float o1 = __shfl_xor(P1[v], off, 32);
            float o2 = __shfl_xor(P2[v], off, 32);
            int   oi = __shfl_xor(BI[v], off, 32);
            float nM = fmaxf(M[v], oM);
            float e1 = __expf(M[v] - nM), e2 = __expf(oM - nM);
            S[v]  = S[v]  * e1 + oS * e2;
            P0[v] = P0[v] * e1 + o0 * e2;
            P1[v] = P1[v] * e1 + o1 * e2;
            P2[v] = P2[v] * e1 + o2 * e2;
            if (oM > M[v] || (oM == M[v] && oi < BI[v])) BI[v] = oi;
            M[v] = nM;
        }
    }

    if (l15 == 0) {
#pragma unroll
        for (int v = 0; v < 8; ++v) {
            int row = n0 + half * 8 + v;
            float inv = 1.0f / S[v];
            corr[(b * NPTS + row) * 3 + 0] = P0[v] * inv;
            corr[(b * NPTS + row) * 3 + 1] = P1[v] * inv;
            corr[(b * NPTS + row) * 3 + 2] = P2[v] * inv;
            corres_i[b * NPTS + row] = BI[v];
            out[OFF_C + b * NPTS + row] = (float)BI[v];
        }
    }
}

// ---------------------------------------------------------------------------
// Kernel 2: gmcce — kNN(10) on src points + triangle-consistency loss.
// One block per batch; both point clouds cached in LDS (48 KB).
// ---------------------------------------------------------------------------
__global__ __launch_bounds__(256) void gmcce_kernel(
    const float* __restrict__ src, const float* __restrict__ corr,
    float* __restrict__ loss_out, unsigned int* __restrict__ min_out)
{
    __shared__ float sp[NPTS * 3];
    __shared__ float cpt[NPTS * 3];
    const int b = blockIdx.x;
    const float* sb = src + b * 3 * NPTS;
    for (int i = threadIdx.x; i < NPTS; i += 256) {
        sp[i * 3 + 0] = sb[i];
        sp[i * 3 + 1] = sb[NPTS + i];
        sp[i * 3 + 2] = sb[2 * NPTS + i];
        cpt[i * 3 + 0] = corr[(b * NPTS + i) * 3 + 0];
        cpt[i * 3 + 1] = corr[(b * NPTS + i) * 3 + 1];
        cpt[i * 3 + 2] = corr[(b * NPTS + i) * 3 + 2];
    }
    __syncthreads();

    for (int n = threadIdx.x; n < NPTS; n += 256) {
        const float px = sp[n * 3], py = sp[n * 3 + 1], pz = sp[n * 3 + 2];
        float bd[10]; int bi[10];
#pragma unroll
        for (int t = 0; t < 10; ++t) { bd[t] = 3.4e38f; bi[t] = 0; }
        for (int m = 0; m < NPTS; ++m) {
            float dx = px - sp[m * 3], dy = py - sp[m * 3 + 1], dz = pz - sp[m * 3 + 2];
            float d2 = dx * dx + dy * dy + dz * dz;
            if (d2 < 0.1f - 1e-7f) continue;          // reference -inf mask
            if (d2 < bd[9]) {
                bd[9] = d2; bi[9] = m;
#pragma unroll
                for (int t = 9; t >= 1; --t) {
                    if (bd[t] < bd[t - 1]) {
                        float td = bd[t]; bd[t] = bd[t - 1]; bd[t - 1] = td;
                        int   ti = bi[t]; bi[t] = bi[t - 1]; bi[t - 1] = ti;
                    }
                }
            }
        }

        // 45 triangle losses; keep 10 smallest
        float tl[10];
#pragma unroll
        for (int t = 0; t < 10; ++t) tl[t] = 3.4e38f;
        const float qx = cpt[n * 3], qy = cpt[n * 3 + 1], qz = cpt[n * 3 + 2];
        for (int i = 0; i < 9; ++i) {
            const int ia = bi[i];
            const float sax = sp[ia * 3],  say = sp[ia * 3 + 1],  saz = sp[ia * 3 + 2];
            const float cax = cpt[ia * 3], cay = cpt[ia * 3 + 1], caz = cpt[ia * 3 + 2];
            for (int j = i + 1; j < 10; ++j) {
                const int ib = bi[j];
                const float sbx = sp[ib * 3],  sby = sp[ib * 3 + 1],  sbz = sp[ib * 3 + 2];
                const float cbx = cpt[ib * 3], cby = cpt[ib * 3 + 1], cbz = cpt[ib * 3 + 2];
                float d01 = (px - sax) * (px - sax) + (py - say) * (py - say) + (pz - saz) * (pz - saz);
                float d12 = (sax - sbx) * (sax - sbx) + (say - sby) * (say - sby) + (saz - sbz) * (saz - sbz);
                float d02 = (px - sbx) * (px - sbx) + (py - sby) * (py - sby) + (pz - sbz) * (pz - sbz);
                sort3f(d01, d12, d02);
                float e01 = (qx - cax) * (qx - cax) + (qy - cay) * (qy - cay) + (qz - caz) * (qz - caz);
                float e12 = (cax - cbx) * (cax - cbx) + (cay - cby) * (cay - cby) + (caz - cbz) * (caz - cbz);
                float e02 = (qx - cbx) * (qx - cbx) + (qy - cby) * (qy - cby) + (qz - cbz) * (qz - cbz);
                sort3f(e01, e12, e02);
                e01 += 1e-6f; e12 += 1e-6f; e02 += 1e-6f;
                float u0 = d01 - e01, u1 = d12 - e12, u2 = d02 - e02;
                float v0 = d01 + e01, v1 = d12 + e12, v2 = d02 + e02;
                float lp = (u0 * u0 + u1 * u1 + u2 * u2) / (v0 * v0 + v1 * v1 + v2 * v2);
                if (lp < tl[9]) {
                    tl[9] = lp;
#pragma unroll
                    for (int t = 9; t >= 1; --t) {
                        if (tl[t] < tl[t - 1]) { float td = tl[t]; tl[t] = tl[t - 1]; tl[t - 1] = td; }
                    }
                }
            }
        }
        float aco = 0.f;
#pragma unroll
        for (int t = 0; t < 10; ++t) aco += sqrtf(tl[t] + 1e-6f);
        float L = aco * 0.1f;
        loss_out[b * NPTS + n] = L;
        atomicMin(&min_out[b], __float_as_uint(L));   // L >= 0: uint order == float order
    }
}

// ---------------------------------------------------------------------------
// Kernel 3: weight = (2*sigmoid(-30*(loss - min_b)) > 0.6) ? 1 : 0
// ---------------------------------------------------------------------------
__global__ __launch_bounds__(256) void weight_kernel(
    const float* __restrict__ loss_in, const unsigned int* __restrict__ min_in,
    float* __restrict__ out)
{
    int i = blockIdx.x * 256 + threadIdx.x;
    if (i >= BATCH * NPTS) return;
    int b = i >> 11;
    float l = loss_in[i] - __uint_as_float(min_in[b]);
    float w = 2.0f / (1.0f + __expf(30.0f * l));
    out[OFF_W + i] = (w > 0.6f) ? 1.0f : 0.0f;
}

// ---------------------------------------------------------------------------
// Kernel 4: weighted Procrustes per batch (3x3 SVD via Jacobi on S^T S).
// ---------------------------------------------------------------------------
__global__ __launch_bounds__(256) void procrustes_kernel(
    const float* __restrict__ src, const float* __restrict__ tgt,
    const int* __restrict__ corres_i, float* __restrict__ out)
{
    __shared__ float red[256];
    const int b = blockIdx.x;
    const int tid = threadIdx.x;
    const float* sb = src + b * 3 * NPTS;
    const float* tb = tgt + b * 3 * NPTS;
    const float* wp = out + OFF_W + b * NPTS;
    const int*   ci = corres_i + b * NPTS;

    auto reduce = [&](float v) -> float {
        red[tid] = v; __syncthreads();
        for (int s = 128; s > 0; s >>= 1) {
            if (tid < s) red[tid] += red[tid + s];
            __syncthreads();
        }
        float r = red[0]; __syncthreads();
        return r;
    };

    float sw = 0, sx0 = 0, sx1 = 0, sx2 = 0, sy0 = 0, sy1 = 0, sy2 = 0;
    for (int n = tid; n < NPTS; n += 256) {
        float w = wp[n];
        if (w != 0.0f) {
            int c = ci[n];
            sw  += w;
            sx0 += w * sb[n]; sx1 += w * sb[NPTS + n]; sx2 += w * sb[2 * NPTS + n];
            sy0 += w * tb[c]; sy1 += w * tb[NPTS + c]; sy2 += w * tb[2 * NPTS + c];
        }
    }
    float W1  = reduce(sw);
    float inv = 1.0f / (W1 + 1e-7f);
    float mux0 = reduce(sx0) * inv, mux1 = reduce(sx1) * inv, mux2 = reduce(sx2) * inv;
    float muy0 = reduce(sy0) * inv, muy1 = reduce(sy1) * inv, muy2 = reduce(sy2) * inv;

    float m[9] = {0, 0, 0, 0, 0, 0, 0, 0, 0};
    for (int n = tid; n < NPTS; n += 256) {
        float w = wp[n];
        if (w != 0.0f) {
            int c = ci[n];
            float x0 = sb[n] - mux0, x1 = sb[NPTS + n] - mux1, x2 = sb[2 * NPTS + n] - mux2;
            float y0 = tb[c] - muy0, y1 = tb[NPTS + c] - muy1, y2 = tb[2 * NPTS + c] - muy2;
            m[0] += w * y0 * x0; m[1] += w * y0 * x1; m[2] += w * y0 * x2;
            m[3] += w * y1 * x0; m[4] += w * y1 * x1; m[5] += w * y1 * x2;
            m[6] += w * y2 * x0; m[7] += w * y2 * x1; m[8] += w * y2 * x2;
        }
    }
    float A[3][3];
    for (int t = 0; t < 9; ++t) {
        float r = reduce(m[t]);
        A[t / 3][t % 3] = r * inv;
    }

    if (tid == 0) {
        // C = A^T A, cyclic Jacobi -> V (columns = eigenvectors)
        float C[3][3];
        for (int i = 0; i < 3; ++i)
            for (int j = 0; j < 3; ++j)
                C[i][j] = A[0][i] * A[0][j] + A[1][i] * A[1][j] + A[2][i] * A[2][j];
        float V[3][3] = {{1, 0, 0}, {0, 1, 0}, {0, 0, 1}};
        for (int sweep = 0; sweep < 12; ++sweep) {
            for (int pq = 0; pq < 3; ++pq) {
                const int p = (pq == 2) ? 1 : 0;
                const int q = (pq == 0) ? 1 : 2;
                float apq = C[p][q];
                if (fabsf(apq) < 1e-30f) continue;
                float th = (C[q][q] - C[p][p]) / (2.0f * apq);
                float t  = ((th >= 0.f) ? 1.f : -1.f) / (fabsf(th) + sqrtf(th * th + 1.f));
                float cc = rsqrtf(t * t + 1.f);
                float ss = t * cc;
                for (int k = 0; k < 3; ++k) {
                    float cpk = C[p][k], cqk = C[q][k];
                    C[p][k] = cc * cpk - ss * cqk;
                    C[q][k] = ss * cpk + cc * cqk;
                }
                for (int k = 0; k < 3; ++k) {
                    float ckp = C[k][p], ckq = C[k][q];
                    C[k][p] = cc * ckp - ss * ckq;
                    C[k][q] = ss * ckp + cc * ckq;
                }
                for (int k = 0; k < 3; ++k) {
                    float vkp = V[k][p], vkq = V[k][q];
                    V[k][p] = cc * vkp - ss * vkq;
                    V[k][q] = ss * vkp + cc * vkq;
                }
            }
        }
        float ev[3] = {C[0][0], C[1][1], C[2][2]};
        int o0 = 0, o1 = 1, o2 = 2, tswap;
        if (ev[o0] < ev[o1]) { tswap = o0; o0 = o1; o1 = tswap; }
        if (ev[o0] < ev[o2]) { tswap = o0; o0 = o2; o2 = tswap; }
        if (ev[o1] < ev[o2]) { tswap = o1; o1 = o2; o2 = tswap; }
        const int ord[3] = {o0, o1, o2};

        float U[3][3], Vc[3][3];
        for (int k = 0; k < 3; ++k) {
            int kk = ord[k];
            float vx = V[0][kk], vy = V[1][kk], vz = V[2][kk];
            Vc[0][k] = vx; Vc[1][k] = vy; Vc[2][k] = vz;
            float ux = A[0][0] * vx + A[0][1] * vy + A[0][2] * vz;
            float uy = A[1][0] * vx + A[1][1] * vy + A[1][2] * vz;
            float uz = A[2][0] * vx + A[2][1] * vy + A[2][2] * vz;
            float nn = sqrtf(ux * ux + uy * uy + uz * uz);
            if (nn > 1e-20f) {
                float r = 1.f / nn; ux *= r; uy *= r; uz *= r;
            } else if (k == 2) {
                ux = U[1][0] * U[2][1] - U[2][0] * U[1][1];
                uy = U[2][0] * U[0][1] - U[0][0] * U[2][1];
                uz = U[0][0] * U[1][1] - U[1][0] * U[0][1];
                float n2 = sqrtf(ux * ux + uy * uy + uz * uz);
                if (n2 > 1e-20f) { float r = 1.f / n2; ux *= r; uy *= r; uz *= r; }
                else { ux = 0; uy = 0; uz = 1; }
            } else {
                ux = (k == 0) ? 1.f : 0.f; uy = (k == 1) ? 1.f : 0.f; uz = 0.f;
            }
            U[0][k] = ux; U[1][k] = uy; U[2][k] = uz;
        }
        float d = (det3(A) < 0.f) ? -1.f : 1.f;
        float R[3][3];
        for (int i = 0; i < 3; ++i)
            for (int j = 0; j < 3; ++j)
                R[i][j] = U[i][0] * Vc[j][0] + U[i][1] * Vc[j][1] + d * U[i][2] * Vc[j][2];
        float mux[3] = {mux0, mux1, mux2};
        float muy[3] = {muy0, muy1, muy2};
        for (int i = 0; i < 3; ++i) {
            for (int j = 0; j < 3; ++j) out[OFF_R + b * 9 + i * 3 + j] = R[i][j];
            out[OFF_T + b * 3 + i] =
                muy[i] - (R[i][0] * mux[0] + R[i][1] * mux[1] + R[i][2] * mux[2]);
        }
    }
}

// ---------------------------------------------------------------------------
extern "C" void kernel_launch(void* const* d_in, const int* in_sizes, int n_in,
                              void* d_out, int out_size, void* d_ws, size_t ws_size,
                              hipStream_t stream)
{
    const float* srcE = (const float*)d_in[0];   // (B, 512, N)
    const float* tgtE = (const float*)d_in[1];   // (B, 512, N)
    const float* src  = (const float*)d_in[2];   // (B, 3, N)
    const float* tgt  = (const float*)d_in[3];   // (B, 3, N)
    // d_in[4] = temperature (unused by reference), d_in[5] = is_corr (unused)
    float* out = (float*)d_out;

    // workspace carve-up (floats): corr_tgt | corres_int | loss | min
    float*        ws_corr   = (float*)d_ws;                            // B*N*3
    int*          ws_corres = (int*)(ws_corr + BATCH * NPTS * 3);      // B*N
    float*        ws_loss   = (float*)(ws_corres + BATCH * NPTS);      // B*N
    unsigned int* ws_min    = (unsigned int*)(ws_loss + BATCH * NPTS); // B

    init_min_kernel<<<1, 64, 0, stream>>>(ws_min);
    score_argmax_kernel<<<BATCH * (NPTS / 16) / 8, 256, 0, stream>>>(
        srcE, tgtE, tgt, ws_corr, ws_corres, out);
    gmcce_kernel<<<BATCH, 256, 0, stream>>>(src, ws_corr, ws_loss, ws_min);
    weight_kernel<<<(BATCH * NPTS + 255) / 256, 256, 0, stream>>>(ws_loss, ws_min, out);
    procrustes_kernel<<<BATCH, 256, 0, stream>>>(src, tgt, ws_corres, out);
}